// SpecAugment_18940805776172
// MI455X (gfx1250) — compile-verified
//
#include <hip/hip_runtime.h>
#include <stdint.h>

// ---------------------------------------------------------------------------
// SpecAugment for MI455X (gfx1250).
//   out[b,t,f] = x[b,t,f] * mask_t[b,t] * mask_f[b,f]
// Stage 1: reproduce JAX threefry2x32 RNG -> per-batch gap params (64 threads)
// Stage 2: materialize mask_t (64x4096) and mask_f (64x256) in scratch
// Stage 3: bandwidth-bound apply pass: per-wave double-buffered Tensor Data
//          Mover pipeline (tensor_load_to_lds + s_wait_tensorcnt), LDS B128
//          reads, non-temporal B128 stores.
// ---------------------------------------------------------------------------

#define B_ 64
#define T_ 4096
#define F_ 256

typedef __attribute__((ext_vector_type(4))) unsigned int u32x4;
typedef __attribute__((ext_vector_type(8))) int          i32x8;
typedef __attribute__((ext_vector_type(4))) int          i32x4;
typedef __attribute__((ext_vector_type(4))) float        f32x4;

// ----------------------------- threefry2x32 --------------------------------
__device__ __forceinline__ uint32_t rotl32(uint32_t v, int n) {
  return (v << n) | (v >> (32 - n));
}

__device__ __forceinline__ void threefry2x32(uint32_t k0, uint32_t k1,
                                             uint32_t c0, uint32_t c1,
                                             uint32_t& o0, uint32_t& o1) {
  uint32_t ks2 = k0 ^ k1 ^ 0x1BD11BDAu;
  uint32_t x0 = c0 + k0;
  uint32_t x1 = c1 + k1;
#define TF_R(r) { x0 += x1; x1 = rotl32(x1, r); x1 ^= x0; }
  TF_R(13) TF_R(15) TF_R(26) TF_R(6)
  x0 += k1;  x1 += ks2 + 1u;
  TF_R(17) TF_R(29) TF_R(16) TF_R(24)
  x0 += ks2; x1 += k0 + 2u;
  TF_R(13) TF_R(15) TF_R(26) TF_R(6)
  x0 += k0;  x1 += k1 + 3u;
  TF_R(17) TF_R(29) TF_R(16) TF_R(24)
  x0 += k1;  x1 += ks2 + 4u;
  TF_R(13) TF_R(15) TF_R(26) TF_R(6)
  x0 += ks2; x1 += k0 + 5u;
#undef TF_R
  o0 = x0; o1 = x1;
}

// JAX random_bits(key, 32, shape) element `idx` of a flat stream of `total`
// words (total even here): counts = iota(total), c0 = first half, c1 = second.
__device__ __forceinline__ uint32_t jax_bits(uint32_t k0, uint32_t k1,
                                             uint32_t idx, uint32_t total) {
  uint32_t half = total >> 1;
  uint32_t a, b;
  if (idx < half) { threefry2x32(k0, k1, idx, idx + half, a, b); return a; }
  threefry2x32(k0, k1, idx - half, idx, a, b); return b;
}

// jax.random.split(key, 3): bits shape (3,2) from counts 0..5.
__device__ __forceinline__ void jax_split3(uint32_t k0, uint32_t k1,
                                           uint32_t out[6]) {
  uint32_t a0, b0, a1, b1, a2, b2;
  threefry2x32(k0, k1, 0u, 3u, a0, b0);
  threefry2x32(k0, k1, 1u, 4u, a1, b1);
  threefry2x32(k0, k1, 2u, 5u, a2, b2);
  // flat [a0,a1,a2,b0,b1,b2] reshaped (3,2):
  out[0] = a0; out[1] = a1;   // key 0
  out[2] = a2; out[3] = b0;   // key 1
  out[4] = b1; out[5] = b2;   // key 2
}

// One axis of _axis_mask: emit 3 (start,end) pairs (inactive -> (0,0)).
__device__ void axis_params(uint32_t ka0, uint32_t ka1, int b, int axis_len,
                            bool keep, int* outp) {
  uint32_t s6[6];
  jax_split3(ka0, ka1, s6);                    // kn, kl, ks
  uint32_t n = jax_bits(s6[0], s6[1], (uint32_t)b, 64u) & 3u;   // randint 0..3
  uint32_t lens[3];
  bool act[3];
  uint32_t maxlen = 0u;
  for (int g = 0; g < 3; ++g) {
    lens[g] = 5u + jax_bits(s6[2], s6[3], (uint32_t)(b * 3 + g), 192u) % 10u;
    act[g] = ((uint32_t)g < n) && keep;
    if (act[g] && lens[g] > maxlen) maxlen = lens[g];
  }
  uint32_t span = (uint32_t)axis_len - maxlen;                  // >= 242
  for (int g = 0; g < 3; ++g) {
    uint32_t st = jax_bits(s6[4], s6[5], (uint32_t)(b * 3 + g), 192u) % span;
    outp[2 * g]     = act[g] ? (int)st : 0;
    outp[2 * g + 1] = act[g] ? (int)(st + lens[g]) : 0;
  }
}

// ------------------------ Stage 1: per-batch params ------------------------
__global__ void sa_params_kernel(int* __restrict__ pF, int* __restrict__ pT) {
  int b = threadIdx.x;
  if (b >= B_) return;
  uint32_t k6[6];
  jax_split3(0u, 42u, k6);                     // key(42) -> kp, kf, kt
  // keep = uniform(kp,(B,)) < 0.5
  uint32_t ub = jax_bits(k6[0], k6[1], (uint32_t)b, (uint32_t)B_);
  float u = __uint_as_float((ub >> 9) | 0x3F800000u) - 1.0f;
  bool keep = u < 0.5f;
  axis_params(k6[2], k6[3], b, F_, keep, pF + b * 6);
  axis_params(k6[4], k6[5], b, T_, keep, pT + b * 6);
}

// ------------------------- Stage 2: mask arrays ----------------------------
__global__ void sa_mask_kernel(const int* __restrict__ pF,
                               const int* __restrict__ pT,
                               float* __restrict__ mask_f,
                               float* __restrict__ mask_t) {
  int i = blockIdx.x * blockDim.x + threadIdx.x;
  const int nf = B_ * F_;
  if (i < nf) {
    int b = i >> 8, pos = i & (F_ - 1);
    const int* p = pF + b * 6;
    float m = 1.0f;
    #pragma unroll
    for (int g = 0; g < 3; ++g)
      if (pos >= p[2 * g] && pos < p[2 * g + 1]) m = 0.0f;
    mask_f[i] = m;
  } else {
    int j = i - nf;
    if (j < B_ * T_) {
      int b = j >> 12, pos = j & (T_ - 1);
      const int* p = pT + b * 6;
      float m = 1.0f;
      #pragma unroll
      for (int g = 0; g < 3; ++g)
        if (pos >= p[2 * g] && pos < p[2 * g + 1]) m = 0.0f;
      mask_t[j] = m;
    }
  }
}

// ---------------- Stage 3: apply pass via Tensor Data Mover ----------------
// Block = 256 threads = 8 waves. Block covers 128 rows of one batch as
// NTILES=4 tiles of 32 rows, double-buffered (2 x 32KB LDS). Each wave owns a
// 4-row x 256-float slice of every tile; it keeps up to 2 TDM loads in flight
// (s_wait_tensorcnt 1 in steady state), reads its slice via ds_load_b128, and
// writes results with non-temporal global_store_b128 (streaming output must
// not evict the L2-resident masks). TENSORcnt is per-wave & in-order, so no
// block barriers are needed anywhere.
#define TILE_ROWS 32
#define NTILES    4

__device__ __forceinline__ void tdm_issue_4x256(unsigned long long gaddr,
                                                uint32_t lds_addr) {
  u32x4 g0 = {
      1u,                                              // [1:0] count = 1
      lds_addr,                                        // [63:32] lds_addr
      (uint32_t)(gaddr & 0xFFFFFFFFull),               // global_addr lo
      (((uint32_t)(gaddr >> 32)) & 0x01FFFFFFu) | (2u << 30)  // addr hi|type=2
  };
  i32x8 g1 = {
      (int)(2u << 16),           // data_size = 4 bytes (code 2)
      (int)((uint32_t)F_ << 16), // tensor_dim0 low16 = 256
      (int)(4u << 16),           // tensor_dim1 low16 = 4 rows
      (int)((uint32_t)F_ << 16), // tile_dim0 = 256
      4,                         // tile_dim1 = 4
      F_,                        // tensor_dim0_stride = 256 elements
      0, 0
  };
  i32x4 gz4 = { 0, 0, 0, 0 };            // groups 2/3 unused (2-D tensor)
  i32x8 gz8 = { 0, 0, 0, 0, 0, 0, 0, 0 };
  __builtin_amdgcn_tensor_load_to_lds(g0, g1, gz4, gz4, gz8, 0);
}

__global__ void sa_apply_kernel(const float* __restrict__ x,
                                const float* __restrict__ mask_t,
                                const float* __restrict__ mask_f,
                                float* __restrict__ out) {
  __shared__ f32x4 tile[2][2048];               // 2 x 32 rows x 64 f32x4 = 64 KB

  const int b     = blockIdx.x >> 5;            // 32 row-groups per batch
  const int grp   = blockIdx.x & 31;
  const int trow0 = grp * (TILE_ROWS * NTILES); // first row of this block
  const int wid   = __builtin_amdgcn_readfirstlane((int)(threadIdx.x >> 5));
  const int lane  = (int)(threadIdx.x & 31u);

  const unsigned long long xbase = (unsigned long long)(uintptr_t)x;
  const uint32_t lds_base = (uint32_t)(uintptr_t)(&tile[0][0]);
  const uint32_t lds_wave = lds_base + (uint32_t)wid * 4096u;   // wave slice

  // row0 of this wave's slice inside tile i:
  //   r(i) = b*T_ + trow0 + i*TILE_ROWS + wid*4
  const size_t rwave = (size_t)b * T_ + (size_t)trow0 + (size_t)(wid * 4);

  // ---- prologue: fill both buffers ----
  tdm_issue_4x256(xbase + (unsigned long long)(rwave + 0 * TILE_ROWS) * (F_ * 4ull),
                  lds_wave);
  tdm_issue_4x256(xbase + (unsigned long long)(rwave + 1 * TILE_ROWS) * (F_ * 4ull),
                  lds_wave + 32768u);

  const f32x4* mf4 = (const f32x4*)(mask_f + (size_t)b * F_);
  f32x4* out4      = (f32x4*)out;

  #pragma unroll
  for (int i = 0; i < NTILES; ++i) {
    // tile i resident when at most (issued - (i+1)) loads outstanding:
    // steady state 1, final iteration 0. In-order per-wave completion.
    if (i < NTILES - 1) __builtin_amdgcn_s_wait_tensorcnt(1);
    else                __builtin_amdgcn_s_wait_tensorcnt(0);

    const int    buf   = i & 1;
    const size_t rbase = rwave + (size_t)(i * TILE_ROWS);

    #pragma unroll
    for (int k = 0; k < 8; ++k) {
      int q   = k * 32 + lane;                 // 0..255 within wave's subtile
      int row = q >> 6;                        // 0..3
      int col = q & 63;                        // f32x4 column
      float mt = mask_t[rbase + (size_t)row];
      f32x4 mf = mf4[col];
      f32x4 v  = tile[buf][wid * 256 + q];     // ds_load_b128
      f32x4 r  = v * mf * mt;
      __builtin_nontemporal_store(r, &out4[(rbase + (size_t)row) * 64 + (size_t)col]);
    }

    if (i + 2 < NTILES) {
      // all ds reads of this buffer retired before the DMA may overwrite it
      asm volatile("s_wait_dscnt 0x0" ::: "memory");
      tdm_issue_4x256(xbase + (unsigned long long)(rwave + (size_t)((i + 2) * TILE_ROWS)) * (F_ * 4ull),
                      lds_wave + (uint32_t)buf * 32768u);
    }
  }
}

// ------------------------------- launcher ----------------------------------
extern "C" void kernel_launch(void* const* d_in, const int* in_sizes, int n_in,
                              void* d_out, int out_size, void* d_ws, size_t ws_size,
                              hipStream_t stream) {
  (void)in_sizes; (void)n_in; (void)out_size; (void)ws_size;
  const float* x  = (const float*)d_in[0];
  float* out      = (float*)d_out;

  // scratch layout (all 16B aligned):
  //   [0..767]     int params: pF[64][6], pT[64][6]
  //   mask_f: 64*256 floats, mask_t: 64*4096 floats
  int*   pF     = (int*)d_ws;
  int*   pT     = pF + B_ * 6;
  float* mask_f = (float*)d_ws + 768;
  float* mask_t = mask_f + B_ * F_;

  sa_params_kernel<<<1, 64, 0, stream>>>(pF, pT);

  const int nmask = B_ * F_ + B_ * T_;
  sa_mask_kernel<<<(nmask + 255) / 256, 256, 0, stream>>>(pF, pT, mask_f, mask_t);

  // B * T/(32*4) = 64 * 32 = 2048 blocks, 256 threads each
  sa_apply_kernel<<<B_ * (T_ / (TILE_ROWS * NTILES)), 256, 0, stream>>>(
      x, mask_t, mask_f, out);
}